// RoIAlignRotated_74483322847576
// MI455X (gfx1250) — compile-verified
//
#include <hip/hip_runtime.h>
#include <hip/hip_bf16.h>

typedef __attribute__((ext_vector_type(2))) float v2f;
typedef __attribute__((ext_vector_type(8))) float v8f;

#define OUT_H 7
#define OUT_W 7
#define FH 128
#define FW 128
#define NCH 256
#define SPATIAL_SCALE 0.25f

// One block per ROI. 256 threads = 8 waves (wave32). Each wave owns two
// 16-channel chunks; per bin we gather 16 taps x 16 channels into the WMMA
// A-matrix layout and contract against the bilinear-weight B-matrix with a
// chain of four V_WMMA_F32_16X16X4_F32 ops (K = 16 taps total).
__global__ __launch_bounds__(256)
void roi_align_rotated_wmma(const float* __restrict__ features,
                            const float* __restrict__ rois,
                            float* __restrict__ out,
                            int nroi) {
    const int roi = blockIdx.x;
    if (roi >= nroi) return;

    const int tid  = threadIdx.x;
    const int wave = tid >> 5;
    const int lane = tid & 31;
    const bool lo  = (lane < 16);   // lane half selects K-pairs {0,1} vs {2,3}

    // ---- ROI parameters (uniform across block) ----
    const float* rp = rois + (size_t)roi * 6;
    const int   bb    = (int)rp[0];
    const float cx    = rp[1] * SPATIAL_SCALE - 0.5f;
    const float cy    = rp[2] * SPATIAL_SCALE - 0.5f;
    const float rw    = rp[3] * SPATIAL_SCALE;
    const float rh    = rp[4] * SPATIAL_SCALE;
    const float theta = rp[5];
    const float cosT  = cosf(theta);
    const float sinT  = sinf(theta);
    const float bin_h = rh * (1.0f / OUT_H);
    const float bin_w = rw * (1.0f / OUT_W);

    // Per-lane feature row base: channel = c0 + (lane & 15), set per chunk below.
    const int chLane = lane & 15;
    const size_t featBatch = (size_t)bb * NCH * FH * FW;

    for (int bin = 0; bin < OUT_H * OUT_W; ++bin) {
        const float phf = (float)(bin / OUT_W);
        const float pwf = (float)(bin % OUT_W);

        // ---- geometry: 4 samples x 4 taps -> 16 (weight, y, x) triples ----
        float w16[16];
        int   ty[16];
        int   tx[16];
#pragma unroll
        for (int s = 0; s < 4; ++s) {
            const int sy = s >> 1, sx = s & 1;
            const float yy = -rh * 0.5f + (phf + ((float)sy + 0.5f) * 0.5f) * bin_h;
            const float xx = -rw * 0.5f + (pwf + ((float)sx + 0.5f) * 0.5f) * bin_w;
            float y = yy * cosT - xx * sinT + cy;
            float x = yy * sinT + xx * cosT + cx;
            const bool valid = (y >= -1.0f) && (y <= (float)FH) &&
                               (x >= -1.0f) && (x <= (float)FW);
            y = fmaxf(y, 0.0f);
            x = fmaxf(x, 0.0f);
            const float yf = floorf(y);
            const float xf = floorf(x);
            int yl = (int)yf; yl = (yl < FH - 1) ? yl : (FH - 1);
            int xl = (int)xf; xl = (xl < FW - 1) ? xl : (FW - 1);
            const int yh = (yl + 1 < FH - 1) ? (yl + 1) : (FH - 1);
            const int xh = (xl + 1 < FW - 1) ? (xl + 1) : (FW - 1);
            const float ly = (yf >= (float)(FH - 1)) ? 0.0f : (y - yf);
            const float lx = (xf >= (float)(FW - 1)) ? 0.0f : (x - xf);
            const float hy = 1.0f - ly;
            const float hx = 1.0f - lx;
            const float vm = valid ? 0.25f : 0.0f;   // fold 1/(gh*gw) here
            w16[s * 4 + 0] = hy * hx * vm; ty[s * 4 + 0] = yl; tx[s * 4 + 0] = xl;
            w16[s * 4 + 1] = hy * lx * vm; ty[s * 4 + 1] = yl; tx[s * 4 + 1] = xh;
            w16[s * 4 + 2] = ly * hx * vm; ty[s * 4 + 2] = yh; tx[s * 4 + 2] = xl;
            w16[s * 4 + 3] = ly * lx * vm; ty[s * 4 + 3] = yh; tx[s * 4 + 3] = xh;
        }

        // ---- two 16-channel chunks per wave -> all 256 channels per block ----
#pragma unroll
        for (int iter = 0; iter < 2; ++iter) {
            const int c0 = (wave + 8 * iter) * 16;
            const int c  = c0 + chLane;
            const float* fb = features + featBatch + (size_t)c * (FH * FW);

            v8f acc = {};
#pragma unroll
            for (int kk = 0; kk < 16; kk += 4) {
                // A-matrix (16 channels x 4 taps): lane m = channel row m.
                // VGPR0 carries K = kk+{0 | 2}, VGPR1 carries K = kk+{1 | 3}
                // for lanes 0-15 | 16-31.  B uses the same K mapping, so the
                // contraction pairs w[k] with G[k] for every k.
                const int y0 = lo ? ty[kk + 0] : ty[kk + 2];
                const int x0 = lo ? tx[kk + 0] : tx[kk + 2];
                const int y1 = lo ? ty[kk + 1] : ty[kk + 3];
                const int x1 = lo ? tx[kk + 1] : tx[kk + 3];
                const float a0 = fb[y0 * FW + x0];
                const float a1 = fb[y1 * FW + x1];
                const float b0 = lo ? w16[kk + 0] : w16[kk + 2];
                const float b1 = lo ? w16[kk + 1] : w16[kk + 3];
                v2f A = {a0, a1};
                v2f B = {b0, b1};
                acc = __builtin_amdgcn_wmma_f32_16x16x4_f32(
                    /*neg_a=*/false, A, /*neg_b=*/false, B,
                    /*c_mod=*/(short)0, acc, /*reuse_a=*/false, /*reuse_b=*/false);
            }

            // ---- extract: lane n, VGPR j holds D[M = j + 8*(n>=16)][n].
            // All N columns are identical, so lanes 0-7 hold channels 0-7 at
            // acc[lane], lanes 16-23 hold channels 8-15 at acc[lane-16].
            const int sel = lane & 7;
            float r = acc[0];
#pragma unroll
            for (int j = 1; j < 8; ++j) r = (sel == j) ? acc[j] : r;

            const bool active = (lane < 8) || (lane >= 16 && lane < 24);
            const int  ch     = (lane < 8) ? lane : (lane - 8);
            if (active) {
                out[((size_t)roi * NCH + (size_t)(c0 + ch)) * (OUT_H * OUT_W) + bin] = r;
            }
        }
    }
}

extern "C" void kernel_launch(void* const* d_in, const int* in_sizes, int n_in,
                              void* d_out, int out_size, void* d_ws, size_t ws_size,
                              hipStream_t stream) {
    const float* features = (const float*)d_in[0];
    const float* rois     = (const float*)d_in[1];
    float*       out      = (float*)d_out;
    const int nroi = in_sizes[1] / 6;

    dim3 grid(nroi);
    dim3 block(256);
    hipLaunchKernelGGL(roi_align_rotated_wmma, grid, block, 0, stream,
                       features, rois, out, nroi);
}